// LocalGCRNNCell_47691316855461
// MI455X (gfx1250) — compile-verified
//
#include <hip/hip_runtime.h>

// ---------------------------------------------------------------------------
// Per-link GRU cell (grouped GEMM) for MI455X / gfx1250.
//   Per link j: C[32,64] = A_h[32,64] x W_h[64,64]  (x3 gates, fp32 WMMA)
//             + A_x[32, 8] x W_x[ 8,64]            (x3 gates)
//   then GRU pointwise combine.
// Memory-bound: ~742 MB streamed per call -> ~32 us at 23.3 TB/s.
// Uses V_WMMA_F32_16X16X4_F32 (fp32 in == fp32 kept; converting weights
// would only add HBM traffic since each weight is used once).
// ---------------------------------------------------------------------------

typedef float v2f __attribute__((ext_vector_type(2)));
typedef float v8f __attribute__((ext_vector_type(8)));

#define NUM_LINKS 10000
#define NF 8
#define NH 64
#define NB 32
#define AH_STRIDE 65   // 64 + 1 pad (bank-conflict avoidance)
#define AX_STRIDE 9    // 8 + 1 pad

__device__ __forceinline__ v8f wmma_f32(v2f a, v2f b, v8f c) {
  // 8 args: (neg_a, A, neg_b, B, c_mod, C, reuse_a, reuse_b)
  return __builtin_amdgcn_wmma_f32_16x16x4_f32(
      false, a, false, b, (short)0, c, false, false);
}

__device__ __forceinline__ float sigmoidf_fast(float x) {
  return 1.0f / (1.0f + __expf(-x));
}

__launch_bounds__(128, 4)
__global__ void gcrnn_cell_kernel(
    const float* __restrict__ hidden,                              // [32,64,L]
    const float* __restrict__ in_i,                                // [32, 8,L]
    const float* __restrict__ rhW, const float* __restrict__ rhB,  // [L,64,64],[64,L]
    const float* __restrict__ riW, const float* __restrict__ riB,  // [L, 8,64],[64,L]
    const float* __restrict__ uhW, const float* __restrict__ uhB,
    const float* __restrict__ uiW, const float* __restrict__ uiB,
    const float* __restrict__ nhW, const float* __restrict__ nhB,
    const float* __restrict__ niW, const float* __restrict__ niB,
    float* __restrict__ out)                                       // [32,64,L]
{
  const int j    = blockIdx.x;        // link id
  const int tid  = threadIdx.x;       // 0..127 (4 wave32s)
  const int lane = tid & 31;
  const int wave = tid >> 5;          // N-tile: columns [16*wave, 16*wave+16)
  const int n0   = wave * 16;

  __shared__ float Ah[NB * AH_STRIDE];  // hidden_j, row-major [b][i]
  __shared__ float Ax[NB * AX_STRIDE];  // in_j,     row-major [b][i]

  // ---- Stage this link's activations into LDS (strided gather; adjacent-j
  //      workgroups reuse the same cachelines via the 192MB L2). ----
#pragma unroll
  for (int e = 0; e < (NB * NH) / 128; ++e) {
    int idx = e * 128 + tid;              // idx == b*64 + i
    Ah[(idx >> 6) * AH_STRIDE + (idx & 63)] =
        hidden[(size_t)idx * NUM_LINKS + j];
  }
#pragma unroll
  for (int e = 0; e < (NB * NF) / 128; ++e) {
    int idx = e * 128 + tid;              // idx == b*8 + i
    Ax[(idx >> 3) * AX_STRIDE + (idx & 7)] =
        in_i[(size_t)idx * NUM_LINKS + j];
  }
  __syncthreads();

  // Per-link weight bases (contiguous 16KB / 2KB blocks -> streaming loads).
  const float* Wr  = rhW + (size_t)j * NH * NH;
  const float* Wu  = uhW + (size_t)j * NH * NH;
  const float* Wn  = nhW + (size_t)j * NH * NH;
  const float* Wxr = riW + (size_t)j * NF * NH;
  const float* Wxu = uiW + (size_t)j * NF * NH;
  const float* Wxn = niW + (size_t)j * NF * NH;

  // WMMA 16x16x4 fp32 fragment lane decomposition:
  //  A (16x4): lanes 0-15 hold rows M=0..15 at K = k,k+1 ; lanes 16-31 at K = k+2,k+3
  //  B (4x16): lanes 0-15 hold row K=k   (N=n0..n0+15)  ; lanes 16-31 row K=k+2
  const int nl = lane & 15;           // column-within-tile / row-within-tile
  const int kh = (lane >> 4) * 2;     // 0 or 2
  const int col = n0 + nl;            // output feature column for this lane

  // Accumulators: reset/update merge hidden+input paths; "new" keeps them
  // separate (reset gate multiplies only the hidden path + its bias).
  v8f aR[2]  = {v8f{}, v8f{}};
  v8f aU[2]  = {v8f{}, v8f{}};
  v8f aNh[2] = {v8f{}, v8f{}};
  v8f aNx[2] = {v8f{}, v8f{}};

  // ---- Hidden GEMMs: K = 64 in steps of 4, 6 WMMAs per step ----
#pragma unroll
  for (int k = 0; k < NH; k += 4) {
    const int kk = k + kh;
    v2f a0, a1, bR, bU, bN;
    a0.x = Ah[nl * AH_STRIDE + kk];         a0.y = Ah[nl * AH_STRIDE + kk + 1];
    a1.x = Ah[(16 + nl) * AH_STRIDE + kk];  a1.y = Ah[(16 + nl) * AH_STRIDE + kk + 1];
    bR.x = Wr[kk * NH + col];  bR.y = Wr[(kk + 1) * NH + col];
    bU.x = Wu[kk * NH + col];  bU.y = Wu[(kk + 1) * NH + col];
    bN.x = Wn[kk * NH + col];  bN.y = Wn[(kk + 1) * NH + col];
    aR[0]  = wmma_f32(a0, bR, aR[0]);   aR[1]  = wmma_f32(a1, bR, aR[1]);
    aU[0]  = wmma_f32(a0, bU, aU[0]);   aU[1]  = wmma_f32(a1, bU, aU[1]);
    aNh[0] = wmma_f32(a0, bN, aNh[0]);  aNh[1] = wmma_f32(a1, bN, aNh[1]);
  }

  // ---- Input GEMMs: K = 8 in steps of 4 ----
#pragma unroll
  for (int k = 0; k < NF; k += 4) {
    const int kk = k + kh;
    v2f a0, a1, bR, bU, bN;
    a0.x = Ax[nl * AX_STRIDE + kk];         a0.y = Ax[nl * AX_STRIDE + kk + 1];
    a1.x = Ax[(16 + nl) * AX_STRIDE + kk];  a1.y = Ax[(16 + nl) * AX_STRIDE + kk + 1];
    bR.x = Wxr[kk * NH + col];  bR.y = Wxr[(kk + 1) * NH + col];
    bU.x = Wxu[kk * NH + col];  bU.y = Wxu[(kk + 1) * NH + col];
    bN.x = Wxn[kk * NH + col];  bN.y = Wxn[(kk + 1) * NH + col];
    aR[0]  = wmma_f32(a0, bR, aR[0]);   aR[1]  = wmma_f32(a1, bR, aR[1]);
    aU[0]  = wmma_f32(a0, bU, aU[0]);   aU[1]  = wmma_f32(a1, bU, aU[1]);
    aNx[0] = wmma_f32(a0, bN, aNx[0]);  aNx[1] = wmma_f32(a1, bN, aNx[1]);
  }

  // ---- Per-lane biases for this output column ----
  const size_t boff = (size_t)col * NUM_LINKS + j;
  const float brc = rhB[boff] + riB[boff];   // reset: hidden+input bias merge
  const float buc = uhB[boff] + uiB[boff];   // update
  const float bnh = nhB[boff];               // new (hidden path, inside reset*)
  const float bni = niB[boff];               // new (input path)

  // ---- GRU pointwise epilogue + strided store ----
  // C layout (16x16 f32): VGPR r, lanes 0-15 -> (M=r, N=lane),
  //                               lanes 16-31 -> (M=r+8, N=lane-16).
#pragma unroll
  for (int m = 0; m < 2; ++m) {
#pragma unroll
    for (int r = 0; r < 8; ++r) {
      const int brow = m * 16 + r + (lane >> 4) * 8;        // batch index b
      const float h  = Ah[brow * AH_STRIDE + col];          // hidden[b,col,j]
      const float rv = sigmoidf_fast(aR[m][r] + brc);
      const float uv = sigmoidf_fast(aU[m][r] + buc);
      const float nv = tanhf(rv * (aNh[m][r] + bnh) + aNx[m][r] + bni);
      out[(size_t)(brow * NH + col) * NUM_LINKS + j] =
          (1.0f - uv) * nv + uv * h;
    }
  }
}

extern "C" void kernel_launch(void* const* d_in, const int* in_sizes, int n_in,
                              void* d_out, int out_size, void* d_ws, size_t ws_size,
                              hipStream_t stream) {
  (void)in_sizes; (void)n_in; (void)out_size; (void)d_ws; (void)ws_size;
  const float* hidden = (const float*)d_in[0];
  const float* in_i   = (const float*)d_in[1];
  const float* rhW    = (const float*)d_in[2];
  const float* rhB    = (const float*)d_in[3];
  const float* riW    = (const float*)d_in[4];
  const float* riB    = (const float*)d_in[5];
  const float* uhW    = (const float*)d_in[6];
  const float* uhB    = (const float*)d_in[7];
  const float* uiW    = (const float*)d_in[8];
  const float* uiB    = (const float*)d_in[9];
  const float* nhW    = (const float*)d_in[10];
  const float* nhB    = (const float*)d_in[11];
  const float* niW    = (const float*)d_in[12];
  const float* niB    = (const float*)d_in[13];
  float* out          = (float*)d_out;

  gcrnn_cell_kernel<<<dim3(NUM_LINKS), dim3(128), 0, stream>>>(
      hidden, in_i,
      rhW, rhB, riW, riB,
      uhW, uhB, uiW, uiB,
      nhW, nhB, niW, niB,
      out);
}